// CZT_prop_11141145166249
// MI455X (gfx1250) — compile-verified
//
#include <hip/hip_runtime.h>
#include <math.h>

// ---------------------------------------------------------------------------
// CZT (dual-pass Bluestein) propagation for MI455X / gfx1250.
// FFT2048 = four-step 16 x 16 x 8; radix-16 stages on V_WMMA_F32_16X16X4_F32,
// radix-8 tail on VALU. Twiddle + DFT16 tables computed once (setup kernel),
// then DMA'd into LDS per block via TENSOR_LOAD_TO_LDS (TDM), together with
// the input row. Wave32, 256-thread (8-wave) blocks.
// Workspace layout (assumed ws_size >= ~64.3 MB):
//   [0,32MB)      bufX   (4 x 1024 x 1024 float2)
//   [32MB,64MB)   bufY
//   [64MB,...)    b0 (4x1024 c), ft (4x2048 c), hpost (4x1024 c),
//                 tTw (2048 c), tD (768 f)
// ---------------------------------------------------------------------------

#define Hn   1024
#define Wn   1024
#define OHn  1024
#define OWn  1024
#define NWL  4
#define DXc  1.0e-4f
#define DYc  1.0e-4f
#define ODXc 1.0e-5f
#define ODYc 1.0e-5f
#define Zc   0.05f
#define TWO_PI_F 6.283185307179586f

typedef float v2f __attribute__((ext_vector_type(2)));
typedef float v8f __attribute__((ext_vector_type(8)));
typedef unsigned int v4u __attribute__((ext_vector_type(4)));
typedef int v4i __attribute__((ext_vector_type(4)));
typedef int v8i __attribute__((ext_vector_type(8)));

static __device__ __forceinline__ v8f wmma4(v2f a, v2f b, v8f c) {
  // D = A(16x4) * B(4x16) + C(16x16), f32.
  return __builtin_amdgcn_wmma_f32_16x16x4_f32(false, a, false, b, (short)0, c,
                                               false, false);
}

// ---------------------------------------------------------------------------
// TDM: 1-D tile load (nelem elements of (1<<dsz_code) bytes) global -> LDS.
// D# per CDNA5 ISA §8: group0 {count=1, lds_addr, global_addr, type=2},
// group1 {data_size, tensor_dim0=nelem, tensor_dim1=1, tile_dim0=nelem,
// tile_dim1=1, dim0_stride=nelem}, groups 2/3 zero (<=2D tensor).
// Issue from ONE wave only; completion via TENSORcnt.
// 6-arg builtin form (clang-23 / therock headers).
// ---------------------------------------------------------------------------
static __device__ __forceinline__ void tdm_load_1d(void* ldsDst,
                                                   const void* gsrc,
                                                   unsigned nelem,
                                                   unsigned dsz_code) {
  unsigned lds_off = (unsigned)(uintptr_t)ldsDst;  // flat addr low 32 = LDS off
  unsigned long long ga = (unsigned long long)(uintptr_t)gsrc;
  v4u g0;
  g0.x = 1u;                                        // count=1, user mode
  g0.y = lds_off;                                   // lds_addr [63:32]
  g0.z = (unsigned)(ga & 0xffffffffull);            // global_addr [95:64]
  g0.w = (unsigned)((ga >> 32) & 0x01ffffffull) | (2u << 30);  // type=2
  v8i g1;
  g1[0] = (int)(dsz_code << 16);  // workgroup_mask=0, data_size
  g1[1] = (int)(nelem << 16);     // tensor_dim0[15:0] @ bits 63:48
  g1[2] = (int)(1u << 16);        // tensor_dim1=1     @ bits 95:80
  g1[3] = (int)(nelem << 16);     // tile_dim0         @ bits 127:112
  g1[4] = 1;                      // tile_dim1=1       @ bits 143:128
  g1[5] = (int)nelem;             // tensor_dim0_stride[31:0]
  g1[6] = 0;
  g1[7] = 0;
  v4i z4 = {0, 0, 0, 0};
  v8i z8 = {0, 0, 0, 0, 0, 0, 0, 0};
  __builtin_amdgcn_tensor_load_to_lds(g0, g1, z4, z4, z8, 0);
}

// DFT16 matrices packed in one 768-float block: Dc | Dp | Dn.
// Dc=cos(2pi kr/16), Dp=+sin, Dn=-sin  (W16^{kr}=cos - i sin)
// Yr = Dc*Xr + Dp*Xi ; Yi = Dc*Xi + Dn*Xr
static __device__ __forceinline__ void fill_dft16(float* D, int tid) {
  int k = tid >> 4, r = tid & 15;
  float ang = -(TWO_PI_F / 16.0f) * (float)(k * r);
  float s, c;
  __sincosf(ang, &s, &c);  // s = -sin(2pi kr/16)
  D[tid] = c;
  D[256 + tid] = -s;
  D[512 + tid] = s;
}

// Twiddle table: Tw[t] = exp(-2*pi*i*t/2048)
static __device__ __forceinline__ void fill_twiddle(float2* Tw, int tid) {
  for (int s8 = 0; s8 < 8; s8++) {
    int j = tid + (s8 << 8);
    float sn, cs;
    __sincosf((-TWO_PI_F / 2048.0f) * (float)j, &sn, &cs);
    Tw[j] = make_float2(cs, sn);
  }
}

// In-LDS 2048-pt FFT. Input in (Xr,Xi) [destroyed], output in (Yr,Yi).
// Natural-order in/out. All 8 waves participate; EXEC all-1s around WMMA.
static __device__ __forceinline__ void fft2048_lds(float* Xr, float* Xi,
                                                   float* Yr, float* Yi,
                                                   const float* D,
                                                   const float2* Tw, int tid) {
  const int lane = tid & 31;
  const int w    = tid >> 5;
  const int mrow = lane & 15;
  const int hi   = (lane >> 4) & 1;  // lanes 16..31
  const int col  = (w << 4) + mrow;  // tile column 0..127
  const float* Dc = D;
  const float* Dp = D + 256;
  const float* Dn = D + 512;

  // Preload A operands (DFT16 chunks) — shared by both WMMA stages.
  v2f Ac[4], Ap[4], An[4];
#pragma unroll
  for (int kk = 0; kk < 4; kk++) {
    int kb = (kk << 2) + (hi ? 2 : 0);
    int ai = (mrow << 4) + kb;
    Ac[kk] = v2f{Dc[ai], Dc[ai + 1]};
    Ap[kk] = v2f{Dp[ai], Dp[ai + 1]};
    An[kk] = v2f{Dn[ai], Dn[ai + 1]};
  }

  // ---- Stage A: 128x DFT16 over r of x[c + 128 r];  X -> Y -------------
  {
    v8f aR = {0.f, 0.f, 0.f, 0.f, 0.f, 0.f, 0.f, 0.f};
    v8f aI = {0.f, 0.f, 0.f, 0.f, 0.f, 0.f, 0.f, 0.f};
#pragma unroll
    for (int kk = 0; kk < 4; kk++) {
      int kb = (kk << 2) + (hi ? 2 : 0);
      int a0 = col + (kb << 7);
      int a1 = col + ((kb + 1) << 7);
      v2f br = {Xr[a0], Xr[a1]};
      v2f bi = {Xi[a0], Xi[a1]};
      aR = wmma4(Ac[kk], br, aR);
      aR = wmma4(Ap[kk], bi, aR);
      aI = wmma4(Ac[kk], bi, aI);
      aI = wmma4(An[kk], br, aI);
    }
#pragma unroll
    for (int j = 0; j < 8; j++) {
      int row = j + (hi ? 8 : 0);  // k1
      float2 t = Tw[row * col];    // W_2048^{c*k1}, product <= 1905
      float vr = aR[j], vi = aI[j];
      Yr[(row << 7) + col] = vr * t.x - vi * t.y;
      Yi[(row << 7) + col] = vr * t.y + vi * t.x;
    }
  }
  __syncthreads();

  // ---- Stage B: inner 128-pt, its own 128x DFT16;  Y -> X --------------
  {
    v8f aR = {0.f, 0.f, 0.f, 0.f, 0.f, 0.f, 0.f, 0.f};
    v8f aI = {0.f, 0.f, 0.f, 0.f, 0.f, 0.f, 0.f, 0.f};
    const int c2    = col & 7;
    const int sbase = ((col >> 3) << 7) + c2;  // k1*128 + c2
#pragma unroll
    for (int kk = 0; kk < 4; kk++) {
      int kb = (kk << 2) + (hi ? 2 : 0);
      int a0 = sbase + (kb << 3);
      int a1 = sbase + ((kb + 1) << 3);
      v2f br = {Yr[a0], Yr[a1]};
      v2f bi = {Yi[a0], Yi[a1]};
      aR = wmma4(Ac[kk], br, aR);
      aR = wmma4(Ap[kk], bi, aR);
      aI = wmma4(Ac[kk], bi, aI);
      aI = wmma4(An[kk], br, aI);
    }
#pragma unroll
    for (int j = 0; j < 8; j++) {
      int row = j + (hi ? 8 : 0);       // k1'
      float2 t = Tw[(c2 * row) << 4];   // 2pi/128 = 16*(2pi/2048)
      float vr = aR[j], vi = aI[j];
      Xr[(row << 7) + col] = vr * t.x - vi * t.y;
      Xi[(row << 7) + col] = vr * t.y + vi * t.x;
    }
  }
  __syncthreads();

  // ---- Stage C: 256x radix-8 over c2;  X -> Y (natural order) ----------
  {
    float zr[8], zi[8];
    const int base = tid << 3;  // = k1'*128 + k1*8
#pragma unroll
    for (int c2 = 0; c2 < 8; c2++) {
      zr[c2] = Xr[base + c2];
      zi[c2] = Xi[base + c2];
    }
    const float R2 = 0.7071067811865476f;
    const float W8r[8] = {1.f, R2, 0.f, -R2, -1.f, -R2, 0.f, R2};
    const float W8i[8] = {0.f, -R2, -1.f, -R2, 0.f, R2, 1.f, R2};
#pragma unroll
    for (int k3 = 0; k3 < 8; k3++) {
      float sr = 0.f, si = 0.f;
#pragma unroll
      for (int c2 = 0; c2 < 8; c2++) {
        int t8 = (c2 * k3) & 7;
        sr += zr[c2] * W8r[t8] - zi[c2] * W8i[t8];
        si += zr[c2] * W8i[t8] + zi[c2] * W8r[t8];
      }
      Yr[tid + (k3 << 8)] = sr;  // k1 + 16 k1' + 256 k3
      Yi[tid + (k3 << 8)] = si;
    }
  }
  __syncthreads();
}

// Rayleigh–Sommerfeld kernel value at (mx,my) for wavelength wlv.
static __device__ __forceinline__ float2 rs_kernel_dev(float mx, float my,
                                                       float wlv) {
  float k  = TWO_PI_F / wlv;
  float r2 = mx * mx + my * my + Zc * Zc;
  float rr = sqrtf(r2);
  float fm = (1.0f / TWO_PI_F) * Zc / r2;
  float s, c;
  __sincosf(k * rr, &s, &c);
  // (c + i s) * (1/rr - i k) * fm
  return make_float2(fm * (c / rr + k * s), fm * (s / rr - k * c));
}

// ---------------------------------------------------------------------------
// Kernel 1: per-wavelength chirp tables b0/ft/hpost (ft via the WMMA FFT),
// plus the shared twiddle table + DFT16 matrices written once (block 0).
// ---------------------------------------------------------------------------
__global__ void czt_setup_kernel(const float* __restrict__ wavelengths,
                                 float2* __restrict__ b0,
                                 float2* __restrict__ ft,
                                 float2* __restrict__ hpost,
                                 float2* __restrict__ tTw,
                                 float* __restrict__ tD) {
  __shared__ float sA[4096];
  __shared__ float sB[4096];
  __shared__ float2 sTw[2048];
  __shared__ float sD[768];
  const int tid = threadIdx.x;
  const int wl  = blockIdx.x;
  const float wlv = wavelengths[wl];
  const float Dm  = wlv * Zc / DXc;
  const float M   = (float)OWn;
  const float f1  = -0.5f * OHn * ODXc + 0.5f * Dm;
  const float f2  =  0.5f * OHn * ODXc + 0.5f * Dm;
  const float dsh = (M * Dm + f2 - f1) / (2.0f * M);
  const float D1  = f1 + dsh;
  const float D2  = f2 + dsh;
  const float alpha_A = TWO_PI_F * D1 / Dm;
  const float alpha_W = -TWO_PI_F * (D1 - D2) / (M * Dm);

  // 1/h (2047 valid, 1 zero pad): 1/h[j] = exp(-i * aW * (j-1023)^2 / 2)
  for (int s8 = 0; s8 < 8; s8++) {
    int j = tid + (s8 << 8);
    float re = 0.f, im = 0.f;
    if (j < 2047) {
      float hk = (float)(j - 1023);
      float ph = 0.5f * alpha_W * hk * hk;
      float sn, cs;
      __sincosf(ph, &sn, &cs);
      re = cs;
      im = -sn;
    }
    sA[j] = re;
    sA[2048 + j] = im;
  }
  fill_dft16(sD, tid);
  fill_twiddle(sTw, tid);
  __syncthreads();

  if (wl == 0) {  // publish shared tables for the row kernels
    for (int s8 = 0; s8 < 8; s8++) {
      int j = tid + (s8 << 8);
      tTw[j] = sTw[j];
    }
    for (int s3 = 0; s3 < 3; s3++) tD[tid + (s3 << 8)] = sD[tid + (s3 << 8)];
  }

  fft2048_lds(sA, sA + 2048, sB, sB + 2048, sD, sTw, tid);

  for (int s8 = 0; s8 < 8; s8++) {
    int j = tid + (s8 << 8);
    ft[(wl << 11) + j] = make_float2(sB[j], sB[2048 + j]);
  }
  // b0[j] = exp(-i aA j) * exp(i aW j^2/2);  hpost = h-slice * m_shift / 2048
  for (int s4 = 0; s4 < 4; s4++) {
    int j = tid + (s4 << 8);
    float fj = (float)j;
    float sn, cs;
    __sincosf(-alpha_A * fj + 0.5f * alpha_W * fj * fj, &sn, &cs);
    b0[(wl << 10) + j] = make_float2(cs, sn);
    float l  = fj / M * (D2 - D1) + D1;
    float ph = 0.5f * alpha_W * fj * fj -
               TWO_PI_F * l * (-(float)Hn * 0.5f + 0.5f) / Dm;
    __sincosf(ph, &sn, &cs);
    const float invN = 1.0f / 2048.0f;
    hpost[(wl << 10) + j] = make_float2(cs * invN, sn * invN);
  }
}

// ---------------------------------------------------------------------------
// Kernel 2: u0 = field * F(input chirp), written TRANSPOSED (LDS-tiled).
// ---------------------------------------------------------------------------
__global__ void czt_pre_kernel(const float* __restrict__ fr,
                               const float* __restrict__ fi,
                               const float* __restrict__ wavelengths,
                               float2* __restrict__ dst) {
  __shared__ float2 tile[32][33];
  const int wl = blockIdx.z;
  const float wlv = wavelengths[wl];
  const int j0 = blockIdx.x << 5;
  const int i0 = blockIdx.y << 5;
  const float step_x = ((float)Hn * DXc) / (float)(Hn - 1);
  const float step_y = ((float)Wn * DYc) / (float)(Wn - 1);
  const float x0 = -0.5f * Hn * DXc;
  const float y0 = -0.5f * Wn * DYc;
#pragma unroll
  for (int rr = 0; rr < 4; rr++) {
    int li = threadIdx.y + (rr << 3);
    int i  = i0 + li;
    int j  = j0 + threadIdx.x;
    float2 F = rs_kernel_dev(x0 + step_x * (float)i, y0 + step_y * (float)j,
                             wlv);
    size_t idx = ((size_t)wl << 20) + ((size_t)i << 10) + j;
    float ur = fr[idx], ui = fi[idx];
    tile[li][threadIdx.x] = make_float2(ur * F.x - ui * F.y,
                                        ur * F.y + ui * F.x);
  }
  __syncthreads();
#pragma unroll
  for (int rr = 0; rr < 4; rr++) {
    int lj = threadIdx.y + (rr << 3);
    dst[((size_t)wl << 20) + ((size_t)(j0 + lj) << 10) + (i0 + threadIdx.x)] =
        tile[threadIdx.x][lj];
  }
}

// ---------------------------------------------------------------------------
// Kernel 3: row-wise Bluestein CZT. TDM pre-stages (row, twiddle, DFT16)
// into LDS, then: x*b0, pad, FFT, *ft, IFFT (conj trick), window, *hpost.
// ---------------------------------------------------------------------------
__global__ void czt_row_kernel(const float2* __restrict__ src,
                               float2* __restrict__ dst,
                               const float2* __restrict__ b0,
                               const float2* __restrict__ ft,
                               const float2* __restrict__ hpost,
                               const float2* __restrict__ tTw,
                               const float* __restrict__ tD) {
  __shared__ float sA[4096];   // Xr | Xi
  __shared__ float sB[4096];   // Yr | Yi ; first 8KB doubles as row staging
  __shared__ float2 sTw[2048];
  __shared__ float sD[768];
  const int tid = threadIdx.x;
  const int row = blockIdx.x;
  const int wl  = blockIdx.y;
  const size_t rbase = ((size_t)wl << 20) + ((size_t)row << 10);

  // Wave 0 issues three TDM DMAs; TENSORcnt tracks completion.
  if (tid < 32) {
    tdm_load_1d(sB, src + rbase, 1024, 3);   // row: 1024 x 8B
    tdm_load_1d(sTw, tTw, 2048, 3);          // twiddle: 2048 x 8B
    tdm_load_1d(sD, tD, 768, 2);             // DFT16 mats: 768 x 4B
    __builtin_amdgcn_s_wait_tensorcnt(0);
  }
  __syncthreads();

  // x * b0, zero-pad to 2048 (staging consumed before fft overwrites sB)
  const float2* stage = (const float2*)sB;
  for (int s4 = 0; s4 < 4; s4++) {
    int j = tid + (s4 << 8);
    float2 u = stage[j];
    float2 b = b0[(wl << 10) + j];
    sA[j]        = u.x * b.x - u.y * b.y;
    sA[2048 + j] = u.x * b.y + u.y * b.x;
    sA[1024 + j]        = 0.f;
    sA[2048 + 1024 + j] = 0.f;
  }
  __syncthreads();

  fft2048_lds(sA, sA + 2048, sB, sB + 2048, sD, sTw, tid);

  // frequency-domain multiply by ft, then conjugate (IFFT via conj trick)
  for (int s8 = 0; s8 < 8; s8++) {
    int j = tid + (s8 << 8);
    float2 f = ft[(wl << 11) + j];
    float pr = sB[j] * f.x - sB[2048 + j] * f.y;
    float pi = sB[j] * f.y + sB[2048 + j] * f.x;
    sA[j] = pr;
    sA[2048 + j] = -pi;
  }
  __syncthreads();

  fft2048_lds(sA, sA + 2048, sB, sB + 2048, sD, sTw, tid);

  for (int s4 = 0; s4 < 4; s4++) {
    int j  = tid + (s4 << 8);
    int jf = 1024 + j;
    float vr = sB[jf];
    float vi = -sB[2048 + jf];  // conj; 1/2048 folded into hpost
    float2 hp = hpost[(wl << 10) + j];
    dst[rbase + j] = make_float2(vr * hp.x - vi * hp.y,
                                 vr * hp.y + vi * hp.x);
  }
}

// ---------------------------------------------------------------------------
// Kernel 4: LDS-tiled complex transpose (between the two CZT passes).
// ---------------------------------------------------------------------------
__global__ void czt_transpose_kernel(const float2* __restrict__ src,
                                     float2* __restrict__ dst) {
  __shared__ float2 tile[32][33];
  const int wl = blockIdx.z;
  const int b0c = blockIdx.x << 5;
  const int a0r = blockIdx.y << 5;
#pragma unroll
  for (int rr = 0; rr < 4; rr++) {
    int la = threadIdx.y + (rr << 3);
    tile[la][threadIdx.x] =
        src[((size_t)wl << 20) + ((size_t)(a0r + la) << 10) +
            (b0c + threadIdx.x)];
  }
  __syncthreads();
#pragma unroll
  for (int rr = 0; rr < 4; rr++) {
    int lb = threadIdx.y + (rr << 3);
    dst[((size_t)wl << 20) + ((size_t)(b0c + lb) << 10) +
        (a0r + threadIdx.x)] = tile[threadIdx.x][lb];
  }
}

// ---------------------------------------------------------------------------
// Kernel 5: out = F0 * v * (Z*ODX*ODY*wl), interleaved complex floats.
// ---------------------------------------------------------------------------
__global__ void czt_post_kernel(const float2* __restrict__ src,
                                const float* __restrict__ wavelengths,
                                float2* __restrict__ out) {
  int g = blockIdx.x * blockDim.x + threadIdx.x;
  if (g >= NWL * Hn * Wn) return;
  const int wl  = g >> 20;
  const int rem = g & ((1 << 20) - 1);
  const int ki  = rem >> 10;
  const int kj  = rem & 1023;
  const float wlv = wavelengths[wl];
  const float step_ox = ((float)OHn * ODXc) / (float)(OHn - 1);
  const float step_oy = ((float)OWn * ODYc) / (float)(OWn - 1);
  float xo = -0.5f * OHn * ODXc + step_ox * (float)ki;
  float yo = -0.5f * OWn * ODYc + step_oy * (float)kj;
  float2 F0 = rs_kernel_dev(xo, yo, wlv);
  float sc  = Zc * ODXc * ODYc * wlv;
  float2 v  = src[g];
  out[g] = make_float2(sc * (v.x * F0.x - v.y * F0.y),
                       sc * (v.x * F0.y + v.y * F0.x));
}

// ---------------------------------------------------------------------------
extern "C" void kernel_launch(void* const* d_in, const int* in_sizes, int n_in,
                              void* d_out, int out_size, void* d_ws,
                              size_t ws_size, hipStream_t stream) {
  (void)in_sizes; (void)n_in; (void)out_size; (void)ws_size;
  const float* fr  = (const float*)d_in[0];
  const float* fi  = (const float*)d_in[1];
  const float* wlp = (const float*)d_in[2];
  float2* out = (float2*)d_out;

  char* ws = (char*)d_ws;
  const size_t fieldBytes = (size_t)NWL * Hn * Wn * sizeof(float2);  // 32 MB
  float2* bufX = (float2*)ws;
  float2* bufY = (float2*)(ws + fieldBytes);
  float2* tb0  = (float2*)(ws + 2 * fieldBytes);
  float2* tft  = tb0 + (size_t)NWL * 1024;
  float2* thp  = tft + (size_t)NWL * 2048;
  float2* tTw  = thp + (size_t)NWL * 1024;
  float*  tD   = (float*)(tTw + 2048);

  czt_setup_kernel<<<dim3(NWL), dim3(256), 0, stream>>>(wlp, tb0, tft, thp,
                                                        tTw, tD);
  czt_pre_kernel<<<dim3(Wn / 32, Hn / 32, NWL), dim3(32, 8), 0, stream>>>(
      fr, fi, wlp, bufX);
  czt_row_kernel<<<dim3(Hn, NWL), dim3(256), 0, stream>>>(bufX, bufY, tb0,
                                                          tft, thp, tTw, tD);
  czt_transpose_kernel<<<dim3(32, 32, NWL), dim3(32, 8), 0, stream>>>(bufY,
                                                                      bufX);
  czt_row_kernel<<<dim3(Hn, NWL), dim3(256), 0, stream>>>(bufX, bufY, tb0,
                                                          tft, thp, tTw, tD);
  czt_post_kernel<<<dim3((NWL * Hn * Wn) / 256), dim3(256), 0, stream>>>(
      bufY, wlp, out);
}